// BiLSTM_Encoder_1580547974588
// MI455X (gfx1250) — compile-verified
//
#include <hip/hip_runtime.h>

typedef float v2f __attribute__((ext_vector_type(2)));
typedef float v8f __attribute__((ext_vector_type(8)));

constexpr int NB  = 64;    // batch
constexpr int SEQ = 512;   // sequence length
constexpr int DIN = 512;   // input dim
constexpr int HID = 512;   // hidden dim
constexpr int NG  = 8;     // gate matrices (4 fwd + 4 bwd)

__device__ __forceinline__ float sigmoidf_(float x) {
    return 1.0f / (1.0f + __expf(-x));
}

// ---------------------------------------------------------------------------
// Kernel 1: input projection  xh[g][s][n][h] = sum_d x[n][s][d]*w_ih[g][d][h] + bias[g][h]
// GEMM: (NB*SEQ, DIN) x (DIN, NG*HID). Block tile 64x128, BK=32.
// 8 waves, each wave owns a 32x32 register tile (2x2 WMMA tiles) -> A/B frag reuse.
// LDS layouts are K-contiguous with even stride so fragments are single b64 loads.
// ---------------------------------------------------------------------------
__global__ __launch_bounds__(256) void proj_gemm(
    const float* __restrict__ x,      // [NB][SEQ][DIN]
    const float* __restrict__ w_ih,   // [NG][DIN][HID]
    const float* __restrict__ bias,   // [NG][HID]
    float* __restrict__ xh)           // [NG][SEQ][NB][HID]
{
    constexpr int BK  = 32;
    constexpr int LDK = BK + 2;            // even stride -> 8B-aligned b64 frags
    __shared__ float As [64][LDK];         // A rows,  K-contiguous
    __shared__ float Bst[128][LDK];        // B cols,  K-contiguous (transposed)

    const int tid  = threadIdx.x;
    const int lane = tid & 31;
    const int wv   = tid >> 5;

    const int rowBase = blockIdx.y * 64;   // over NB*SEQ
    const int colBase = blockIdx.x * 128;  // over NG*HID

    const int m0 = (wv >> 2) * 32;         // wave's 32-row block (0 or 32)
    const int n0 = (wv & 3) * 32;          // wave's 32-col block (0..96)

    const int koff = (lane >> 4) * 2;      // lanes 16-31 hold K=2,3
    const int l15  = lane & 15;

    v8f acc00 = {}, acc01 = {}, acc10 = {}, acc11 = {};

    for (int k0 = 0; k0 < DIN; k0 += BK) {
        // cooperative load A tile (64 x 32), K-contiguous rows
        #pragma unroll
        for (int idx = tid; idx < 64 * BK; idx += 256) {
            int i = idx >> 5;
            int j = idx & 31;
            As[i][j] = x[(size_t)(rowBase + i) * DIN + k0 + j];
        }
        // cooperative load B tile (32 x 128) transposed into [col][k]
        #pragma unroll
        for (int idx = tid; idx < BK * 128; idx += 256) {
            int k = idx >> 7;              // 0..31
            int j = idx & 127;             // col within block (contiguous h: coalesced)
            int c = colBase + j;
            Bst[j][k] = w_ih[((size_t)(c >> 9) * DIN + (k0 + k)) * HID + (c & 511)];
        }
        __syncthreads();

        #pragma unroll
        for (int kk = 0; kk < BK; kk += 4) {
            v2f a0 = *(const v2f*)&As [m0 +      l15][kk + koff];
            v2f a1 = *(const v2f*)&As [m0 + 16 + l15][kk + koff];
            v2f b0 = *(const v2f*)&Bst[n0 +      l15][kk + koff];
            v2f b1 = *(const v2f*)&Bst[n0 + 16 + l15][kk + koff];
            acc00 = __builtin_amdgcn_wmma_f32_16x16x4_f32(false, a0, false, b0,
                                                          (short)0, acc00, false, false);
            acc01 = __builtin_amdgcn_wmma_f32_16x16x4_f32(false, a0, false, b1,
                                                          (short)0, acc01, false, false);
            acc10 = __builtin_amdgcn_wmma_f32_16x16x4_f32(false, a1, false, b0,
                                                          (short)0, acc10, false, false);
            acc11 = __builtin_amdgcn_wmma_f32_16x16x4_f32(false, a1, false, b1,
                                                          (short)0, acc11, false, false);
        }
        __syncthreads();
    }

    // store with bias folded in; xh layout [g][s][n][h]
    #pragma unroll
    for (int msub = 0; msub < 2; ++msub) {
        #pragma unroll
        for (int nsub = 0; nsub < 2; ++nsub) {
            const v8f& acc = msub ? (nsub ? acc11 : acc10) : (nsub ? acc01 : acc00);
            int c = colBase + n0 + nsub * 16 + l15;
            int g = c >> 9, h = c & 511;
            float bv = bias[g * HID + h];
            #pragma unroll
            for (int r = 0; r < 8; ++r) {
                int mrow = (lane < 16) ? r : (r + 8);
                int grow = rowBase + m0 + msub * 16 + mrow;  // flat (n, s)
                int n = grow >> 9;                           // / SEQ
                int s = grow & 511;
                xh[(((size_t)g * SEQ + s) * NB + n) * HID + h] = acc[r] + bv;
            }
        }
    }
}

// ---------------------------------------------------------------------------
// Kernel 2: one recurrence timestep, both directions (blockIdx.z).
// Block owns 32 batch rows x 32 hidden cols across all 4 gates.
// 8 waves = 4 gates x 2 hidden sub-tiles; each wave does 2 batch sub-tiles
// reusing its B fragment. Hidden state double-buffered by step parity.
// ---------------------------------------------------------------------------
__global__ __launch_bounds__(256) void lstm_step(
    const float* __restrict__ xh,     // [NG][SEQ][NB][HID]  (bias included)
    const float* __restrict__ w_hh,   // [NG][HID][HID]
    const float* __restrict__ mask,   // [NB][SEQ]
    float* __restrict__ hbuf,         // [2 dir][2 parity][NB][HID]
    float* __restrict__ cbuf,         // [2 dir][NB][HID]
    float* __restrict__ out_hs,       // [NB][SEQ][2*HID]
    int t)
{
    constexpr int BK  = 32;
    constexpr int LDK = BK + 2;
    __shared__ float As [32][LDK];         // h rows, K-contiguous
    __shared__ float Bst[4][32][LDK];      // per-gate B cols, K-contiguous
    __shared__ float pre[4][32][32];       // gate pre-activations

    const int dir   = blockIdx.z;
    const int u     = dir ? (SEQ - 1 - t) : t;   // original time index consumed
    const int gbase = dir * 4;
    const int nBase = blockIdx.y * 32;
    const int hBase = blockIdx.x * 32;

    const float* hprev = hbuf + ((size_t)dir * 2 + (t & 1)) * NB * HID;
    float*       hnext = hbuf + ((size_t)dir * 2 + ((t + 1) & 1)) * NB * HID;
    float*       cdir  = cbuf + (size_t)dir * NB * HID;

    const int tid  = threadIdx.x;
    const int lane = tid & 31;
    const int wv   = tid >> 5;
    const int gi   = wv >> 1;         // gate 0..3
    const int nsub = wv & 1;          // hidden sub-tile
    const int koff = (lane >> 4) * 2;
    const int l15  = lane & 15;

    v8f acc0 = {}, acc1 = {};

    for (int k0 = 0; k0 < HID; k0 += BK) {
        for (int idx = tid; idx < 32 * BK; idx += 256) {
            int i = idx >> 5;
            int j = idx & 31;
            As[i][j] = hprev[(nBase + i) * HID + k0 + j];
        }
        for (int idx = tid; idx < 4 * BK * 32; idx += 256) {
            int g   = idx >> 10;
            int rem = idx & 1023;
            int k   = rem >> 5;
            int j   = rem & 31;       // contiguous h: coalesced global read
            Bst[g][j][k] = w_hh[(((size_t)(gbase + g)) * HID + (k0 + k)) * HID + hBase + j];
        }
        __syncthreads();

        #pragma unroll
        for (int kk = 0; kk < BK; kk += 4) {
            v2f a0 = *(const v2f*)&As[     l15][kk + koff];
            v2f a1 = *(const v2f*)&As[16 + l15][kk + koff];
            v2f b  = *(const v2f*)&Bst[gi][nsub * 16 + l15][kk + koff];
            acc0 = __builtin_amdgcn_wmma_f32_16x16x4_f32(false, a0, false, b,
                                                         (short)0, acc0, false, false);
            acc1 = __builtin_amdgcn_wmma_f32_16x16x4_f32(false, a1, false, b,
                                                         (short)0, acc1, false, false);
        }
        __syncthreads();
    }

    // stage pre-activations: pre = xh_t + h @ w_hh
    {
        int lc = nsub * 16 + l15;
        int h  = hBase + lc;
        const float* xg = xh + (((size_t)(gbase + gi) * SEQ + u) * NB) * HID;
        #pragma unroll
        for (int r = 0; r < 8; ++r) {
            int mrow = (lane < 16) ? r : (r + 8);
            pre[gi][mrow][lc]      = acc0[r] + xg[(size_t)(nBase + mrow)      * HID + h];
            pre[gi][16 + mrow][lc] = acc1[r] + xg[(size_t)(nBase + 16 + mrow) * HID + h];
        }
    }
    __syncthreads();

    // fused gate nonlinearity + cell/hidden update + masked output
    for (int e = tid; e < 32 * 32; e += 256) {
        int lr = e >> 5;
        int lc = e & 31;
        int n  = nBase + lr;
        int h  = hBase + lc;
        float fg = sigmoidf_(pre[0][lr][lc]);
        float ig = sigmoidf_(pre[1][lr][lc]);
        float og = sigmoidf_(pre[2][lr][lc]);
        float cd = tanhf(pre[3][lr][lc]);
        size_t idx = (size_t)n * HID + h;
        float cnew = fg * cdir[idx] + ig * cd;
        cdir[idx] = cnew;                    // cell updated unconditionally (ref semantics)
        float hnew = og * tanhf(cnew);
        float m = mask[n * SEQ + u];
        float hout = (m == 0.0f) ? hprev[idx] : hnew;
        hnext[idx] = hout;
        out_hs[((size_t)n * SEQ + u) * (2 * HID) + dir * HID + h] = hout * m;
    }
}

// ---------------------------------------------------------------------------
// init / finalize
// ---------------------------------------------------------------------------
__global__ void init_state(const float* __restrict__ h0f, const float* __restrict__ h0b,
                           const float* __restrict__ c0f, const float* __restrict__ c0b,
                           float* __restrict__ hbuf, float* __restrict__ cbuf)
{
    int i = blockIdx.x * blockDim.x + threadIdx.x;   // over 2*NB*HID
    if (i >= 2 * NB * HID) return;
    int dir = i / (NB * HID);
    int rem = i % (NB * HID);
    int h = rem % HID;
    hbuf[(size_t)dir * 2 * NB * HID + rem] = dir ? h0b[h] : h0f[h];  // parity 0
    cbuf[i] = dir ? c0b[h] : c0f[h];
}

__global__ void write_final(const float* __restrict__ hbuf, float* __restrict__ out)
{
    int i = blockIdx.x * blockDim.x + threadIdx.x;   // over 2*NB*HID
    if (i >= 2 * NB * HID) return;
    int dir = i / (NB * HID);
    int rem = i % (NB * HID);
    int n = rem / HID;
    int h = rem % HID;
    // SEQ even -> final state lives at parity 0
    out[(size_t)NB * SEQ * 2 * HID + (size_t)n * 2 * HID + dir * HID + h] =
        hbuf[(size_t)dir * 2 * NB * HID + rem];
}

// ---------------------------------------------------------------------------
extern "C" void kernel_launch(void* const* d_in, const int* in_sizes, int n_in,
                              void* d_out, int out_size, void* d_ws, size_t ws_size,
                              hipStream_t stream)
{
    const float* x    = (const float*)d_in[0];
    const float* mask = (const float*)d_in[1];
    const float* w_ih = (const float*)d_in[2];
    const float* w_hh = (const float*)d_in[3];
    const float* bias = (const float*)d_in[4];
    const float* h0f  = (const float*)d_in[5];
    const float* h0b  = (const float*)d_in[6];
    const float* c0f  = (const float*)d_in[7];
    const float* c0b  = (const float*)d_in[8];
    float* out = (float*)d_out;

    // workspace layout
    float* xh = (float*)d_ws;                          // NG*SEQ*NB*HID
    size_t xh_elems = (size_t)NG * SEQ * NB * HID;
    float* hbuf = xh + xh_elems;                       // 2*2*NB*HID
    float* cbuf = hbuf + (size_t)2 * 2 * NB * HID;     // 2*NB*HID

    init_state<<<dim3((2 * NB * HID + 255) / 256), dim3(256), 0, stream>>>(
        h0f, h0b, c0f, c0b, hbuf, cbuf);

    proj_gemm<<<dim3((NG * HID) / 128, (NB * SEQ) / 64), dim3(256), 0, stream>>>(
        x, w_ih, bias, xh);

    for (int t = 0; t < SEQ; ++t) {
        lstm_step<<<dim3(HID / 32, NB / 32, 2), dim3(256), 0, stream>>>(
            xh, w_hh, mask, hbuf, cbuf, out, t);
    }

    write_final<<<dim3((2 * NB * HID + 255) / 256), dim3(256), 0, stream>>>(hbuf, out);
}